// NCELoss_53111565582366
// MI455X (gfx1250) — compile-verified
//
#include <hip/hip_runtime.h>

typedef float f32x4 __attribute__((ext_vector_type(4)));
typedef float v2f   __attribute__((ext_vector_type(2)));
typedef float v8f   __attribute__((ext_vector_type(8)));

#define NROWS 16384
#define NCOLS 4096
#define LOG2E 1.4426950408889634f
#define LN2   0.6931471805599453f
#define NEG_BIG (-3.402823466e38f)

// Stage 1: one wave32 per row. Streaming NT float4 loads of labels+logits,
// online base-2 logsumexp + argmax(labels) with the scaled logit captured at
// the argmax position. Wave shuffle reduction, lane 0 writes per-row loss.
__global__ __launch_bounds__(256) void nce_rowpass(
    const float* __restrict__ labels, const float* __restrict__ logits,
    const float* __restrict__ alpha, float* __restrict__ partial)
{
  const int lane = threadIdx.x & 31;
  const int wave = threadIdx.x >> 5;
  const int row  = blockIdx.x * 8 + wave;
  const size_t rowoff = (size_t)row * NCOLS;
  const f32x4* lab = (const f32x4*)(labels + rowoff);
  const f32x4* lgt = (const f32x4*)(logits + rowoff);
  const float k = LOG2E / alpha[0];   // y = logit * k  =>  scaled*log2(e)

  float m = NEG_BIG, s = 0.0f;        // running max / sum of 2^(y-m)
  float blab = NEG_BIG, bpos = 0.0f;  // best label value, y at that column
  int   bidx = 0x7fffffff;

  #pragma unroll 4
  for (int j = 0; j < 32; ++j) {
    const int f = j * 32 + lane;                    // float4 index in row
    f32x4 a = __builtin_nontemporal_load(lab + f);
    f32x4 x = __builtin_nontemporal_load(lgt + f);
    const float x0 = x.x * k, x1 = x.y * k, x2 = x.z * k, x3 = x.w * k;
    const int c0 = f * 4;
    // first-occurrence argmax: strict >, ascending scan order per lane
    if (a.x > blab) { blab = a.x; bidx = c0;     bpos = x0; }
    if (a.y > blab) { blab = a.y; bidx = c0 + 1; bpos = x1; }
    if (a.z > blab) { blab = a.z; bidx = c0 + 2; bpos = x2; }
    if (a.w > blab) { blab = a.w; bidx = c0 + 3; bpos = x3; }
    // online logsumexp (base 2): 5 v_exp_f32 per 4 elements, branch-free
    const float m4 = fmaxf(fmaxf(x0, x1), fmaxf(x2, x3));
    const float nm = fmaxf(m, m4);
    s = s * __builtin_amdgcn_exp2f(m - nm)
      + __builtin_amdgcn_exp2f(x0 - nm) + __builtin_amdgcn_exp2f(x1 - nm)
      + __builtin_amdgcn_exp2f(x2 - nm) + __builtin_amdgcn_exp2f(x3 - nm);
    m = nm;
  }

  // wave32 reduction: merge (m,s) pairs and argmax triples
  for (int off = 16; off > 0; off >>= 1) {
    const float om = __shfl_xor(m, off, 32);
    const float os = __shfl_xor(s, off, 32);
    const float nm = fmaxf(m, om);
    s = s * __builtin_amdgcn_exp2f(m - nm) + os * __builtin_amdgcn_exp2f(om - nm);
    m = nm;
    const float ol = __shfl_xor(blab, off, 32);
    const float op = __shfl_xor(bpos, off, 32);
    const int   oi = __shfl_xor(bidx, off, 32);
    if (ol > blab || (ol == blab && oi < bidx)) { blab = ol; bidx = oi; bpos = op; }
  }

  if (lane == 0) {
    // loss = ln(sum e^scaled) - scaled[pos] = ln2 * (m + log2(s) - y_pos)
    partial[row] = LN2 * (m + __builtin_amdgcn_logf(s) - bpos);
  }
}

// Stage 2: deterministic mean of 16384 per-row losses using
// V_WMMA_F32_16X16X4_F32 as an f32 tensor-core accumulator: D = A*ones + C.
// Every A element is summed into all 16 columns of its D row, so the grand
// total over D's 256 entries is 16x the true sum.
__global__ __launch_bounds__(256) void nce_reduce(
    const float* __restrict__ partial, float* __restrict__ out)
{
  __shared__ float wsum[8];
  const int lane = threadIdx.x & 31;
  const int wave = threadIdx.x >> 5;

  v2f ones; ones.x = 1.0f; ones.y = 1.0f;   // B = 4x16 all-ones (2 VGPRs)
  v8f c = {};                                // C/D accumulator (8 VGPRs)
  const float* base = partial + wave * 2048; // each wave sums 2048 values

  #pragma unroll 4
  for (int i = 0; i < 32; ++i) {
    v2f a;                                   // A = 64 f32 across the wave
    a.x = base[i * 64 + lane];
    a.y = base[i * 64 + 32 + lane];
    // 8 args: (neg_a, A, neg_b, B, c_mod, C, reuse_a, reuse_b)
    c = __builtin_amdgcn_wmma_f32_16x16x4_f32(
        false, a, false, ones, (short)0, c, false, false);
  }

  float t = c[0] + c[1] + c[2] + c[3] + c[4] + c[5] + c[6] + c[7];
  for (int off = 16; off > 0; off >>= 1) t += __shfl_xor(t, off, 32);
  if (lane == 0) wsum[wave] = t;
  __syncthreads();

  if (threadIdx.x == 0) {
    float total = 0.0f;
    for (int w = 0; w < 8; ++w) total += wsum[w];
    out[0] = total / (16.0f * (float)NROWS);  // undo 16x replication + mean
  }
}

extern "C" void kernel_launch(void* const* d_in, const int* in_sizes, int n_in,
                              void* d_out, int out_size, void* d_ws, size_t ws_size,
                              hipStream_t stream) {
  const float* labels = (const float*)d_in[0];
  const float* logits = (const float*)d_in[1];
  // d_in[2] = mask (unused by the reference)
  const float* alpha  = (const float*)d_in[3];
  float* partial = (float*)d_ws;   // 16384 floats = 64 KB scratch

  nce_rowpass<<<NROWS / 8, 256, 0, stream>>>(labels, logits, alpha, partial);
  nce_reduce<<<1, 256, 0, stream>>>(partial, (float*)d_out);
}